// YOLOX_69552700391791
// MI455X (gfx1250) — compile-verified
//
#include <hip/hip_runtime.h>
#include <stdint.h>

// ---------------------------------------------------------------------------
// YOLOX post-processing for MI455X (gfx1250, wave32).
// Pipeline: zero-scratch -> per-level score+histogram -> WMMA-assisted cutoff
// scan -> compaction (recompute, no 43MB score spill) -> per-level bitonic
// top-1000 + box decode -> single-block sort + greedy class-aware NMS.
// Workspace use (d_ws): ~232 KB.
// ---------------------------------------------------------------------------

#define NCLS   80
#define TOPK   1000
#define CONFT  0.05f
#define NMST   0.6f
#define NBINS  4096
#define BIN_BASE 12160        // 0x3F80 - 4096: bins of (float_bits>>16); top 4096 bins
#define CAP    4096           // candidate capacity per level
#define NDET   3000
#define SORTN  4096

typedef __attribute__((ext_vector_type(2))) float v2f;
typedef __attribute__((ext_vector_type(8))) float v8f;
typedef unsigned long long u64;
typedef unsigned int u32;

__device__ __forceinline__ float sigm(float x) { return 1.0f / (1.0f + __expf(-x)); }

__device__ __forceinline__ int score_bin(float s) {
    int r = (int)(__float_as_uint(s) >> 16) - BIN_BASE;
    return r < 0 ? 0 : r;
}

// -------------------------------- init ------------------------------------
__global__ void zero_u32(u32* p, int n) {
    int i = blockIdx.x * blockDim.x + threadIdx.x;
    if (i < n) p[i] = 0u;
}

// --------------------- pass 1: fused score + histogram ---------------------
__global__ void score_hist(const float* __restrict__ obj, const float* __restrict__ cls,
                           u32* __restrict__ hist, int M) {
    __shared__ u32 lh[NBINS];
    for (int i = threadIdx.x; i < NBINS; i += blockDim.x) lh[i] = 0u;
    __syncthreads();
    int gstride = gridDim.x * blockDim.x;
    for (int a = blockIdx.x * blockDim.x + threadIdx.x; a < M; a += gstride) {
        float so = sigm(obj[a]);
        const float4* row = (const float4*)(cls + (size_t)a * NCLS);
        __builtin_prefetch(cls + (size_t)(a + gstride) * NCLS, 0, 1); // global_prefetch_b8
        #pragma unroll
        for (int c = 0; c < NCLS / 4; ++c) {
            float4 v = row[c];
            float s0 = sqrtf(so * sigm(v.x));
            float s1 = sqrtf(so * sigm(v.y));
            float s2 = sqrtf(so * sigm(v.z));
            float s3 = sqrtf(so * sigm(v.w));
            atomicAdd(&lh[score_bin(s0)], 1u);
            atomicAdd(&lh[score_bin(s1)], 1u);
            atomicAdd(&lh[score_bin(s2)], 1u);
            atomicAdd(&lh[score_bin(s3)], 1u);
        }
    }
    __syncthreads();
    for (int i = threadIdx.x; i < NBINS; i += blockDim.x) {
        u32 v = lh[i];
        if (v) atomicAdd(&hist[i], v);
    }
}

// ---------------- pass 2: cutoff scan (WMMA segment sums) ------------------
// 4096 bins -> 256 segment sums of 16 contiguous bins each, computed with
// v_wmma_f32_16x16x4_f32 chains. B = all-ones (layout independent);
// A 16x4 f32 layout per ISA: lane L holds M=L&15; VGPR0 = K (L<16?0:2),
// VGPR1 = K+1. D row n==0 read at lane 0 (M=0..7) / lane 16 (M=8..15).
__global__ void scan_cutoff(const u32* __restrict__ hist_all, u32* __restrict__ cutoff) {
    __shared__ float fh[NBINS];
    __shared__ float seg[256];
    int lvl = blockIdx.x;
    const u32* hist = hist_all + lvl * NBINS;
    for (int i = threadIdx.x; i < NBINS; i += blockDim.x) fh[i] = (float)hist[i];
    __syncthreads();

    int wave = threadIdx.x >> 5, lane = threadIdx.x & 31;
    int m = lane & 15, kk = (lane < 16) ? 0 : 2;
    v2f ones = {1.0f, 1.0f};
    for (int g = wave * 2; g < wave * 2 + 2; ++g) {     // 16 groups x 16 segments
        v8f acc = {0.f, 0.f, 0.f, 0.f, 0.f, 0.f, 0.f, 0.f};
        int base = g * 256;
        #pragma unroll
        for (int c = 0; c < 4; ++c) {
            v2f a;
            a.x = fh[base + m * 16 + c * 4 + kk];
            a.y = fh[base + m * 16 + c * 4 + kk + 1];
            acc = __builtin_amdgcn_wmma_f32_16x16x4_f32(
                false, a, false, ones, (short)0, acc, false, false);
        }
        if (lane == 0)  { for (int r = 0; r < 8; ++r) seg[g * 16 + r]     = acc[r]; }
        if (lane == 16) { for (int r = 0; r < 8; ++r) seg[g * 16 + 8 + r] = acc[r]; }
    }
    __syncthreads();

    if (threadIdx.x == 0) {
        float run = 0.0f;
        int s;
        for (s = 255; s >= 0; --s) {                 // suffix scan over segments
            if (run + seg[s] >= (float)TOPK) break;
            run += seg[s];
        }
        int cut = 0;
        if (s >= 0) {
            int b;
            for (b = 15; b >= 0; --b) {              // refine within segment
                if (run + fh[s * 16 + b] >= (float)TOPK) break;
                run += fh[s * 16 + b];
            }
            if (b < 0) b = 0;
            cut = s * 16 + b;
        }
        cutoff[lvl] = (u32)cut;
    }
}

// ------------------- pass 3: compaction (recompute scores) -----------------
__global__ void compact(const float* __restrict__ obj, const float* __restrict__ cls,
                        const u32* __restrict__ cutp, u64* __restrict__ cand,
                        u32* __restrict__ cnt, int M) {
    u32 cut = *cutp;
    int gstride = gridDim.x * blockDim.x;
    for (int a = blockIdx.x * blockDim.x + threadIdx.x; a < M; a += gstride) {
        float so = sigm(obj[a]);
        const float4* row = (const float4*)(cls + (size_t)a * NCLS);
        __builtin_prefetch(cls + (size_t)(a + gstride) * NCLS, 0, 1);
        #pragma unroll
        for (int c = 0; c < NCLS / 4; ++c) {
            float4 v = row[c];
            float sc[4] = {sigm(v.x), sigm(v.y), sigm(v.z), sigm(v.w)};
            #pragma unroll
            for (int q = 0; q < 4; ++q) {
                float s = sqrtf(so * sc[q]);
                if ((u32)score_bin(s) >= cut) {
                    u32 pos = atomicAdd(cnt, 1u);
                    if (pos < CAP) {
                        u32 idx = (u32)a * NCLS + (u32)(c * 4 + q);
                        // key desc, then lower idx first (stable top_k ties)
                        cand[pos] = ((u64)__float_as_uint(s) << 32) | (0xFFFFFFFFu - idx);
                    }
                }
            }
        }
    }
}

// ----------------------------- bitonic (desc) ------------------------------
__device__ void bitonic_desc(u64* s, int N) {
    for (int k = 2; k <= N; k <<= 1) {
        for (int j = k >> 1; j > 0; j >>= 1) {
            __syncthreads();
            for (int t = threadIdx.x; t < N; t += blockDim.x) {
                int ixj = t ^ j;
                if (ixj > t) {
                    u64 a = s[t], b = s[ixj];
                    bool sw = ((t & k) == 0) ? (a < b) : (a > b);
                    if (sw) { s[t] = b; s[ixj] = a; }
                }
            }
        }
    }
    __syncthreads();
}

// ---------------- pass 4: per-level top-1000 + box decode ------------------
__global__ void sort_select(const u64* __restrict__ cand, const u32* __restrict__ cntp,
                            const float* __restrict__ reg,
                            float* __restrict__ det_box, float* __restrict__ det_score,
                            int* __restrict__ det_label, u32* __restrict__ det_valid,
                            int slot_base, int W, float stride) {
    __shared__ u64 sk[CAP];
    u32 count = *cntp;
    if (count > CAP) count = CAP;
    for (int i = threadIdx.x; i < CAP; i += blockDim.x)
        sk[i] = (i < (int)count) ? cand[i] : 0ull;
    __syncthreads();
    bitonic_desc(sk, CAP);

    for (int t = threadIdx.x; t < TOPK; t += blockDim.x) {
        int slot = slot_base + t;
        if (t < (int)count) {
            u64 c = sk[t];
            u32 key = (u32)(c >> 32);
            u32 idx = 0xFFFFFFFFu - (u32)(c & 0xFFFFFFFFull);
            float s = __uint_as_float(key);
            int anc = (int)(idx / NCLS), lab = (int)(idx % NCLS);
            float4 r = ((const float4*)reg)[anc];
            float ax = ((float)(anc % W) + 0.5f) * stride;
            float ay = ((float)(anc / W) + 0.5f) * stride;
            float cx = ax + r.x * stride, cy = ay + r.y * stride;
            float hw = __expf(r.z) * stride * 0.5f, hh = __expf(r.w) * stride * 0.5f;
            det_box[slot * 4 + 0] = cx - hw;
            det_box[slot * 4 + 1] = cy - hh;
            det_box[slot * 4 + 2] = cx + hw;
            det_box[slot * 4 + 3] = cy + hh;
            det_score[slot] = s;
            det_label[slot] = lab;
            det_valid[slot] = (s > CONFT) ? 1u : 0u;
        } else {
            det_box[slot * 4 + 0] = 0.f; det_box[slot * 4 + 1] = 0.f;
            det_box[slot * 4 + 2] = 0.f; det_box[slot * 4 + 3] = 0.f;
            det_score[slot] = 0.f; det_label[slot] = 0; det_valid[slot] = 0u;
        }
    }
}

// ---------------- pass 5: global sort + greedy class-aware NMS -------------
__global__ void nms_kernel(const float* __restrict__ det_box, const float* __restrict__ det_score,
                           const int* __restrict__ det_label, const u32* __restrict__ det_valid,
                           float* __restrict__ out) {
    __shared__ u64 sk[SORTN];
    __shared__ u32 skeep[NDET];
    for (int i = threadIdx.x; i < SORTN; i += blockDim.x) {
        if (i < NDET)
            sk[i] = ((u64)__float_as_uint(det_score[i]) << 32) | (0xFFFFFFFFu - (u32)i);
        else
            sk[i] = 0ull;
    }
    __syncthreads();
    bitonic_desc(sk, SORTN);

    for (int p = threadIdx.x; p < NDET; p += blockDim.x) {
        u32 g = 0xFFFFFFFFu - (u32)(sk[p] & 0xFFFFFFFFull);
        skeep[p] = det_valid[g];
    }
    __syncthreads();

    // Greedy suppression. Label equality replaces the +8192*label offset
    // trick (mathematically identical: offset boxes of different classes
    // never intersect).
    for (int i = 0; i < NDET; ++i) {
        if (skeep[i]) {
            u32 gi = 0xFFFFFFFFu - (u32)(sk[i] & 0xFFFFFFFFull);
            float4 bi = ((const float4*)det_box)[gi];
            int li = det_label[gi];
            float ai = (bi.z - bi.x) * (bi.w - bi.y);
            for (int j = i + 1 + (int)threadIdx.x; j < NDET; j += blockDim.x) {
                if (!skeep[j]) continue;
                u32 gj = 0xFFFFFFFFu - (u32)(sk[j] & 0xFFFFFFFFull);
                if (det_label[gj] != li) continue;
                float4 bj = ((const float4*)det_box)[gj];
                float lx = fmaxf(bi.x, bj.x), ly = fmaxf(bi.y, bj.y);
                float rx = fminf(bi.z, bj.z), ry = fminf(bi.w, bj.w);
                float iw = fmaxf(rx - lx, 0.f), ih = fmaxf(ry - ly, 0.f);
                float inter = iw * ih;
                float aj = (bj.z - bj.x) * (bj.w - bj.y);
                float iou = inter / (ai + aj - inter + 1e-9f);
                if (iou > NMST) skeep[j] = 0u;
            }
        }
        __syncthreads();
    }

    // outputs: boxes[3000*4], scores[3000], labels[3000], keep[3000]
    for (int p = threadIdx.x; p < NDET; p += blockDim.x) {
        u32 g = 0xFFFFFFFFu - (u32)(sk[p] & 0xFFFFFFFFull);
        float4 b = ((const float4*)det_box)[g];
        out[p * 4 + 0] = b.x; out[p * 4 + 1] = b.y;
        out[p * 4 + 2] = b.z; out[p * 4 + 3] = b.w;
        out[NDET * 4 + p] = det_score[g];
        out[NDET * 5 + p] = (float)det_label[g];
        out[NDET * 6 + p] = skeep[p] ? 1.0f : 0.0f;
    }
}

// ------------------------------- launcher ----------------------------------
extern "C" void kernel_launch(void* const* d_in, const int* in_sizes, int n_in,
                              void* d_out, int out_size, void* d_ws, size_t ws_size,
                              hipStream_t stream) {
    (void)n_in; (void)out_size; (void)ws_size;
    const float* obj[3] = {(const float*)d_in[0], (const float*)d_in[3], (const float*)d_in[6]};
    const float* cls[3] = {(const float*)d_in[1], (const float*)d_in[4], (const float*)d_in[7]};
    const float* reg[3] = {(const float*)d_in[2], (const float*)d_in[5], (const float*)d_in[8]};
    int M[3] = {in_sizes[0], in_sizes[3], in_sizes[6]};
    int Wd[3] = {320, 160, 80};
    float strd[3] = {8.0f, 16.0f, 32.0f};

    // workspace layout
    char* ws = (char*)d_ws;
    u32* hist      = (u32*)ws;                      // 3*4096 u32 = 49152 B
    u32* cutoff    = (u32*)(ws + 49152);            // 3 u32
    u32* cnt       = (u32*)(ws + 49152 + 16);       // 3 u32 (padded)
    u64* cand      = (u64*)(ws + 49216);            // 3*CAP u64 = 98304 B
    float* det_box   = (float*)(ws + 147520);       // 3000*4 f32 = 48000 B
    float* det_score = (float*)(ws + 195520);       // 3000 f32
    int*   det_label = (int*)  (ws + 207520);       // 3000 i32
    u32*   det_valid = (u32*)  (ws + 219520);       // 3000 u32  (end 231520)

    zero_u32<<<(12304 + 255) / 256, 256, 0, stream>>>((u32*)d_ws, 12304);

    for (int l = 0; l < 3; ++l) {
        int nb = (M[l] + 255) / 256;
        score_hist<<<nb, 256, 0, stream>>>(obj[l], cls[l], hist + l * NBINS, M[l]);
    }
    scan_cutoff<<<3, 256, 0, stream>>>(hist, cutoff);
    for (int l = 0; l < 3; ++l) {
        int nb = (M[l] + 255) / 256;
        compact<<<nb, 256, 0, stream>>>(obj[l], cls[l], cutoff + l,
                                        cand + (size_t)l * CAP, cnt + l, M[l]);
    }
    for (int l = 0; l < 3; ++l) {
        sort_select<<<1, 1024, 0, stream>>>(cand + (size_t)l * CAP, cnt + l, reg[l],
                                            det_box, det_score, det_label, det_valid,
                                            l * TOPK, Wd[l], strd[l]);
    }
    nms_kernel<<<1, 1024, 0, stream>>>(det_box, det_score, det_label, det_valid,
                                       (float*)d_out);
}